// Energy_20409684591408
// MI455X (gfx1250) — compile-verified
//
#include <hip/hip_runtime.h>
#include <hip/hip_bf16.h>

// ---------------- problem constants (match reference) ----------------
#define NMOL   256
#define MS     48
#define NORB   192              // 4*MS
#define NATOMS (NMOL*MS)        // 12288
#define ZEFF   4.0f
#define NSWEEP 8                // fixed Jacobi sweeps (f32, random dense symmetric)

static constexpr size_t MAT   = (size_t)NORB * NORB;        // 36864
static constexpr size_t OFF_F = 0;
static constexpr size_t OFF_E = (size_t)NMOL * MAT;          // 9437184
static constexpr size_t OFF_P = OFF_E + (size_t)NMOL * NORB; // 9486336
static constexpr size_t OFF_H = OFF_P + (size_t)NMOL * MAT;  // 18923520
static constexpr size_t OFF_W = OFF_H + (size_t)NMOL * MAT;  // 28360704

typedef __attribute__((ext_vector_type(2))) float v2f;
typedef __attribute__((ext_vector_type(8))) float v8f;

// ---------------- trivial helpers ----------------
__global__ void zero_kernel(float* __restrict__ p, long n) {
    long i = (long)blockIdx.x * 256 + threadIdx.x;
    if (i < n) p[i] = 0.f;
}

// One thread per atom pair: w, W-matrix, Hcore off-diagonal 4x4 blocks, vcore atomics.
__global__ void pair_kernel(const int* __restrict__ idxi, const int* __restrict__ idxj,
                            const float* __restrict__ rij, const float* __restrict__ gss,
                            const float* __restrict__ beta, const float* __restrict__ alpha,
                            float* __restrict__ w_out, float* __restrict__ Wmat,
                            float* __restrict__ Hcore, float* __restrict__ vcore, int n_pairs) {
    int p = blockIdx.x * 256 + threadIdx.x;
    if (p >= n_pairs) return;
    int i = idxi[p], j = idxj[p];
    float r = rij[p];
    float rho = 0.5f * (1.f / gss[i] + 1.f / gss[j]);
    float w = rsqrtf(r * r + rho * rho);
    w_out[p] = w;
    atomicAdd(&vcore[i], -ZEFF * w);
    atomicAdd(&vcore[j], -ZEFF * w);
    int mol = i / MS, il = i - mol * MS, jl = j - (j / MS) * MS;
    Wmat[(size_t)mol * MS * MS + il * MS + jl] = w;
    Wmat[(size_t)mol * MS * MS + jl * MS + il] = w;
    float S = expf(-0.5f * (alpha[i] + alpha[j]) * r);
    float bi0 = beta[2 * i], bi1 = beta[2 * i + 1];
    float bj0 = beta[2 * j], bj1 = beta[2 * j + 1];
    float* H = Hcore + (size_t)mol * MAT;
    for (int a = 0; a < 4; ++a) {
        float ba = a ? bi1 : bi0;
        for (int b = 0; b < 4; ++b) {
            float bb = b ? bj1 : bj0;
            float off = 0.5f * (ba + bb) * S;
            H[(size_t)(il * 4 + a) * NORB + (jl * 4 + b)] = off;   // (i,j) block
            H[(size_t)(jl * 4 + b) * NORB + (il * 4 + a)] = off;   // transpose block
        }
    }
}

// Diagonal one-center terms: H[o][o] = uss/upp + vcore
__global__ void diag_kernel(const float* __restrict__ uss, const float* __restrict__ upp,
                            const float* __restrict__ vcore, float* __restrict__ Hcore) {
    int idx = blockIdx.x * 256 + threadIdx.x;
    if (idx >= NATOMS * 4) return;
    int atom = idx >> 2, a = idx & 3;
    int mol = atom / MS, il = atom - mol * MS;
    float u = (a == 0 ? uss[atom] : upp[atom]) + vcore[atom];
    int o = il * 4 + a;
    Hcore[(size_t)mol * MAT + (size_t)o * NORB + o] = u;
}

// Atomic populations q = tr(P diag block); also zeros pot for the next kernel.
__global__ void q_kernel(const float* __restrict__ P, float* __restrict__ qv,
                         float* __restrict__ pot) {
    int a = blockIdx.x * 256 + threadIdx.x;
    if (a >= NATOMS) return;
    int mol = a / MS, il = a - mol * MS;
    const float* Pm = P + (size_t)mol * MAT;
    float s = 0.f;
    for (int x = 0; x < 4; ++x) { int o = il * 4 + x; s += Pm[(size_t)o * NORB + o]; }
    qv[a] = s;
    pot[a] = 0.f;
}

__global__ void pot_kernel(const int* __restrict__ idxi, const int* __restrict__ idxj,
                           const float* __restrict__ w, const float* __restrict__ qv,
                           float* __restrict__ pot, int n_pairs) {
    int p = blockIdx.x * 256 + threadIdx.x;
    if (p >= n_pairs) return;
    int i = idxi[p], j = idxj[p];
    float ww = w[p];
    atomicAdd(&pot[i], ww * qv[j]);
    atomicAdd(&pot[j], ww * qv[i]);
}

// F = Hcore + diag(pot per orbital) - 0.5 * W(i,j) * P   (off-diagonal atom blocks)
__global__ void buildF_kernel(const float* __restrict__ H, const float* __restrict__ pot,
                              const float* __restrict__ Wmat, const float* __restrict__ P,
                              float* __restrict__ F) {
    long idx = (long)blockIdx.x * 256 + threadIdx.x;
    if (idx >= (long)NMOL * (long)MAT) return;
    int mol = (int)(idx / (long)MAT);
    int rem = (int)(idx - (long)mol * (long)MAT);
    int o1 = rem / NORB, o2 = rem - o1 * NORB;
    int i = o1 >> 2, j = o2 >> 2;
    float f = H[idx];
    if (o1 == o2) f += pot[mol * MS + i];
    if (i != j) f -= 0.5f * Wmat[(size_t)mol * MS * MS + i * MS + j] * P[idx];
    F[idx] = f;
}

// ---------------- batched eigh + WMMA density build ----------------
// One molecule per workgroup. Dynamic LDS: A[192*192] + V[192*192] = 294912 B
// (needs CDNA5's 320KB-per-WGP LDS). 8 Jacobi sweeps with round-robin parallel
// ordering (96 disjoint rotations/step), then eigenvalue ranking (ascending),
// compaction of occupied eigenvectors, and P = 2*Vocc*Vocc^T via
// V_WMMA_F32_16X16X4_F32 (f32 in, f32 accumulate).
__global__ __launch_bounds__(256)
void eigh_density_kernel(const float* __restrict__ Fin, float* __restrict__ e_out,
                         float* __restrict__ P, const int* __restrict__ nocc, int mix) {
    extern __shared__ float lds[];
    float* A = lds;          // 192*192, later reused as Vocc[row*96 + slot]
    float* V = lds + MAT;    // 192*192 eigenvectors
    __shared__ float cc[96], ssn[96];
    __shared__ short pp[96], qq[96];
    __shared__ float evals[NORB];
    __shared__ int rnk[NORB];

    const int mol = blockIdx.x;
    const int t = threadIdx.x;
    const float* Fm = Fin + (size_t)mol * MAT;

    // Load A (+ jitter diag) and V = I
    for (int idx = t; idx < (int)MAT; idx += 256) {
        int r = idx / NORB, c = idx - r * NORB;
        float v = Fm[idx];
        if (r == c) v += 1e-6f * (float)r;
        A[idx] = v;
        V[idx] = (r == c) ? 1.f : 0.f;
    }
    __syncthreads();

    for (int sweep = 0; sweep < NSWEEP; ++sweep) {
        for (int step = 0; step < NORB - 1; ++step) {
            if (t < 96) {
                // round-robin tournament: 96 disjoint pairs covering all indices
                int p = (t == 0) ? 0 : 1 + (t - 1 + step) % (NORB - 1);
                int q = 1 + (NORB - 2 - t + step) % (NORB - 1);
                if (p > q) { int z = p; p = q; q = z; }
                float app = A[p * NORB + p], aqq = A[q * NORB + q], apq = A[p * NORB + q];
                float c = 1.f, s = 0.f;
                if (fabsf(apq) > 1e-12f) {
                    float tau = (aqq - app) / (2.f * apq);
                    float tt = 1.f / (fabsf(tau) + sqrtf(1.f + tau * tau));
                    if (tau < 0.f) tt = -tt;
                    c = rsqrtf(1.f + tt * tt);
                    s = tt * c;
                }
                pp[t] = (short)p; qq[t] = (short)q; cc[t] = c; ssn[t] = s;
            }
            __syncthreads();
            // row rotations: A <- J^T A
            for (int idx = t; idx < 96 * NORB; idx += 256) {
                int k = idx / NORB, col = idx - k * NORB;
                int p = pp[k], q = qq[k];
                float c = cc[k], s = ssn[k];
                float ap = A[p * NORB + col], aq = A[q * NORB + col];
                A[p * NORB + col] = c * ap - s * aq;
                A[q * NORB + col] = s * ap + c * aq;
            }
            __syncthreads();
            // column rotations: A <- A J ; V <- V J
            for (int idx = t; idx < 96 * NORB; idx += 256) {
                int k = idx / NORB, row = idx - k * NORB;
                int p = pp[k], q = qq[k];
                float c = cc[k], s = ssn[k];
                float ap = A[row * NORB + p], aq = A[row * NORB + q];
                A[row * NORB + p] = c * ap - s * aq;
                A[row * NORB + q] = s * ap + c * aq;
                float vp = V[row * NORB + p], vq = V[row * NORB + q];
                V[row * NORB + p] = c * vp - s * vq;
                V[row * NORB + q] = s * vp + c * vq;
            }
            __syncthreads();
        }
    }

    if (t < NORB) evals[t] = A[t * NORB + t];
    __syncthreads();

    // ascending rank of each eigenvalue (ties broken by index); zero Vocc region
    if (t < NORB) {
        float ev = evals[t];
        int r = 0;
        for (int j2 = 0; j2 < NORB; ++j2) {
            float o = evals[j2];
            r += (o < ev) || (o == ev && j2 < t);
        }
        rnk[t] = r;
    }
    for (int idx = t; idx < NORB * 96; idx += 256) A[idx] = 0.f;  // A reused as Vocc
    __syncthreads();

    const int noccm = min(nocc[mol], 96);
    float* Vocc = A;  // [row*96 + slot], slot = rank of occupied eigenvalue
    for (int idx = t; idx < (int)MAT; idx += 256) {
        int r = idx / NORB, c = idx - r * NORB;
        int rk = rnk[c];
        if (rk < noccm) Vocc[r * 96 + rk] = V[r * NORB + c];
    }
    if (t < NORB) e_out[(size_t)mol * NORB + rnk[t]] = evals[t];
    __syncthreads();

    // P = 2 * Vocc * Vocc^T via WMMA f32 16x16x4; 12x12 tiles, 8 waves.
    const int lane = t & 31, wv = t >> 5;
    const int m = lane & 15, hi = lane >> 4;
    float* Pm = P + (size_t)mol * MAT;
    for (int tile = wv; tile < 144; tile += 8) {
        int ti = tile / 12, tj = tile - ti * 12;
        v8f acc = {0.f, 0.f, 0.f, 0.f, 0.f, 0.f, 0.f, 0.f};
        for (int k = 0; k < 96; k += 4) {
            // A-frag 16x4: lane m holds K = 2*hi, 2*hi+1   (ISA 32-bit A layout)
            v2f a, b;
            a.x = Vocc[(ti * 16 + m) * 96 + k + 2 * hi];
            a.y = Vocc[(ti * 16 + m) * 96 + k + 2 * hi + 1];
            // B-frag 4x16 = Vocc rows of tile tj (B = Vocc^T)
            b.x = Vocc[(tj * 16 + m) * 96 + k + 2 * hi];
            b.y = Vocc[(tj * 16 + m) * 96 + k + 2 * hi + 1];
            acc = __builtin_amdgcn_wmma_f32_16x16x4_f32(
                false, a, false, b, (short)0, acc, false, false);
        }
        for (int r8 = 0; r8 < 8; ++r8) {
            int grow = ti * 16 + r8 + 8 * hi;      // C/D layout: VGPR r -> M=r / M=r+8
            int gcol = tj * 16 + m;
            size_t o = (size_t)grow * NORB + gcol;
            float val = 2.f * acc[r8];
            Pm[o] = mix ? (0.5f * Pm[o] + 0.5f * val) : val;
        }
    }
}

// ---------------- host orchestration ----------------
extern "C" void kernel_launch(void* const* d_in, const int* in_sizes, int n_in,
                              void* d_out, int out_size, void* d_ws, size_t ws_size,
                              hipStream_t stream) {
    const int*   idxi  = (const int*)d_in[0];
    const int*   idxj  = (const int*)d_in[1];
    // d_in[2]=maskd, d_in[3]=mask: block-index bookkeeping, not needed (dense layout)
    const float* rij   = (const float*)d_in[4];
    const float* uss   = (const float*)d_in[5];
    const float* upp   = (const float*)d_in[6];
    const float* gss   = (const float*)d_in[7];
    const float* beta  = (const float*)d_in[8];
    const float* alpha = (const float*)d_in[9];
    const int*   nocc  = (const int*)d_in[10];
    const int n_pairs = in_sizes[4];

    float* out = (float*)d_out;
    float* F = out + OFF_F;
    float* e = out + OFF_E;
    float* P = out + OFF_P;
    float* H = out + OFF_H;
    float* w = out + OFF_W;

    float* ws    = (float*)d_ws;
    float* vcore = ws;                                   // NATOMS
    float* Wmat  = vcore + NATOMS;                       // NMOL*MS*MS
    float* qv    = Wmat + (size_t)NMOL * MS * MS;        // NATOMS
    float* pot   = qv + NATOMS;                          // NATOMS

    const size_t ldsBytes = 2 * MAT * sizeof(float);     // 294912 B (CDNA5 WGP LDS)
    hipFuncSetAttribute(reinterpret_cast<const void*>(eigh_density_kernel),
                        hipFuncAttributeMaxDynamicSharedMemorySize, (int)ldsBytes);

    const long nH = (long)NMOL * (long)MAT;
    zero_kernel<<<(unsigned)((nH + 255) / 256), 256, 0, stream>>>(H, nH);
    zero_kernel<<<(NATOMS + 255) / 256, 256, 0, stream>>>(vcore, NATOMS);

    pair_kernel<<<(n_pairs + 255) / 256, 256, 0, stream>>>(
        idxi, idxj, rij, gss, beta, alpha, w, Wmat, H, vcore, n_pairs);
    diag_kernel<<<(NATOMS * 4 + 255) / 256, 256, 0, stream>>>(uss, upp, vcore, H);

    // e, v, P = density(Hcore)
    eigh_density_kernel<<<NMOL, 256, ldsBytes, stream>>>(H, e, P, nocc, 0);

    // SCF: 6 fixed iterations with 0.5/0.5 linear mixing
    for (int it = 0; it < 6; ++it) {
        q_kernel<<<(NATOMS + 255) / 256, 256, 0, stream>>>(P, qv, pot);
        pot_kernel<<<(n_pairs + 255) / 256, 256, 0, stream>>>(idxi, idxj, w, qv, pot, n_pairs);
        buildF_kernel<<<(unsigned)((nH + 255) / 256), 256, 0, stream>>>(H, pot, Wmat, P, F);
        eigh_density_kernel<<<NMOL, 256, ldsBytes, stream>>>(F, e, P, nocc, 1);
    }
    // d_out now holds: F (last fock), e (last density), P (mixed), Hcore, w.
}